// CETMAE_project_late_bart_9938554323213
// MI455X (gfx1250) — compile-verified
//
#include <hip/hip_runtime.h>

// Problem constants (from reference setup_inputs): N=64, L=512, D=840.
#define N_  64
#define L_  512
#define D_  840
#define D4_ 210   // D_/4 float4 chunks per row (3360 B, 16B aligned)

// ---- CDNA5 async global<->LDS data movers (ASYNCcnt path) -------------------
#if __has_builtin(__builtin_amdgcn_global_load_async_to_lds_b128) && \
    __has_builtin(__builtin_amdgcn_global_store_async_from_lds_b128)
#define USE_ASYNC_LDS 1
#else
#define USE_ASYNC_LDS 0
#endif

// Operand types expected by the async builtins (from hipcc diagnostics):
//   arg0: "__vector_size__(4*sizeof(int)) int __device__ *"  -> v4i addrspace(1)*
//   arg1: "__vector_size__(4*sizeof(int)) int __shared__ *"  -> v4i addrspace(3)*
typedef int v4i __attribute__((vector_size(4 * sizeof(int))));
#define GV4I(p) ((__attribute__((address_space(1))) v4i*)(p))
#define LV4I(p) ((__attribute__((address_space(3))) v4i*)(p))

__device__ __forceinline__ void async_wait0() {
#if __has_builtin(__builtin_amdgcn_s_wait_asynccnt)
  __builtin_amdgcn_s_wait_asynccnt(0);
#else
  asm volatile("s_wait_asynccnt 0" ::: "memory");
#endif
}

// -----------------------------------------------------------------------------
// Kernel 1: per-sample keep/permutation + the three [N,L] masks.
// One block per sample, L_ threads. Stable O(L^2) rank in LDS replaces argsort.
// -----------------------------------------------------------------------------
__global__ void __launch_bounds__(L_) cetmae_mask_perm(
    const float* __restrict__ noise, const int* __restrict__ lengths,
    int* __restrict__ perm_ws, int* __restrict__ lk_ws,
    float* __restrict__ out) {
  const int n = blockIdx.x;
  const int i = threadIdx.x;

  __shared__ float s_noise[L_];
  __shared__ int   s_keep[L_];

  int len = lengths[n];
  len = len < 16 ? 16 : (len > L_ ? L_ : len);   // jnp.clip(lengths, 16, L)
  const int last = len - 1;                      // last attended pos (always masked)
  const int lk   = len >> 1;                     // floor(len * (1-0.5))

  s_noise[i] = noise[n * L_ + i];
  __syncthreads();

  // rank of i among candidates (stable: ties broken by original index)
  int keep = 0;
  if (i < last) {
    const float ni = s_noise[i];
    int rank = 0;
    for (int j = 0; j < last; ++j) {
      const float nj = s_noise[j];
      rank += (nj < ni) || ((nj == ni) && (j < i));
    }
    keep = (rank < lk) ? 1 : 0;   // exactly lk positions kept (lk <= last)
  }
  s_keep[i] = keep;
  __syncthreads();

  // exclusive prefix count of kept positions before i
  int kb = 0;
  for (int j = 0; j < i; ++j) kb += s_keep[j];

  // argsort of (keep ? pos : pos+L): kept ascending first, then rest ascending
  const int dest = keep ? kb : (lk + (i - kb));
  perm_ws[n * L_ + dest] = i;
  if (i == 0) lk_ws[n] = lk;

  // masks live after masked_x in the flat output
  float* mask = out  + (size_t)N_ * L_ * D_;
  float* inv  = mask + (size_t)N_ * L_;
  float* rem  = inv  + (size_t)N_ * L_;
  const float m = (i < lk) ? 1.0f : 0.0f;
  mask[n * L_ + i] = m;
  inv [n * L_ + i] = 1.0f - m;
  rem [n * L_ + i] = ((!keep) && (i < len)) ? 1.0f : 0.0f;
}

// -----------------------------------------------------------------------------
// Kernel 2: row gather. One block per output row (t,n). Kept rows are streamed
// global -> LDS -> global with per-lane async B128 transfers (CDNA5 ASYNCcnt
// data movers); tail rows (t >= len_keep) are zero-filled (mask multiply).
// -----------------------------------------------------------------------------
__global__ void __launch_bounds__(256) cetmae_gather(
    const float* __restrict__ x, const int* __restrict__ perm_ws,
    const int* __restrict__ lk_ws, float* __restrict__ out) {
  const int t = blockIdx.x;   // output row within sample
  const int n = blockIdx.y;   // sample
  const int c = threadIdx.x;  // float4 chunk id

  __shared__ float4 buf[D4_];
  (void)buf;

  const int lk = lk_ws[n];
  float4* dst = (float4*)out + (size_t)(n * L_ + t) * D4_;

  if (t < lk) {
    const int src = perm_ws[n * L_ + t];
    const float4* srow = (const float4*)x + (size_t)(n * L_ + src) * D4_;
    if (c < D4_) {
#if USE_ASYNC_LDS
      __builtin_amdgcn_global_load_async_to_lds_b128(
          GV4I(srow + c), LV4I(&buf[c]), 0, 0);
      async_wait0();   // LDS data valid for this wave's own chunks
      __builtin_amdgcn_global_store_async_from_lds_b128(
          GV4I(dst + c), LV4I(&buf[c]), 0, 0);
      async_wait0();   // drain before block retires / LDS reuse
#else
      dst[c] = srow[c];
#endif
    }
  } else if (c < D4_) {
    dst[c] = make_float4(0.f, 0.f, 0.f, 0.f);
  }
}

// -----------------------------------------------------------------------------
extern "C" void kernel_launch(void* const* d_in, const int* in_sizes, int n_in,
                              void* d_out, int out_size, void* d_ws, size_t ws_size,
                              hipStream_t stream) {
  (void)in_sizes; (void)n_in; (void)out_size; (void)ws_size;
  const float* x       = (const float*)d_in[0];   // [N, L, D] f32
  const float* noise   = (const float*)d_in[1];   // [N, L]    f32
  const int*   lengths = (const int*)  d_in[2];   // [N]       i32
  float* out = (float*)d_out;                     // masked_x | mask | inv | removed

  int* perm_ws = (int*)d_ws;            // N*L ints
  int* lk_ws   = perm_ws + N_ * L_;     // N ints

  cetmae_mask_perm<<<dim3(N_), dim3(L_), 0, stream>>>(noise, lengths, perm_ws, lk_ws, out);
  cetmae_gather  <<<dim3(L_, N_), dim3(256), 0, stream>>>(x, perm_ws, lk_ws, out);
}